// My_loss_46677704573495
// MI455X (gfx1250) — compile-verified
//
#include <hip/hip_runtime.h>
#include <hip/hip_bf16.h>
#include <math.h>

typedef __attribute__((ext_vector_type(2))) float v2f;
typedef __attribute__((ext_vector_type(8))) float v8f;

#define NBLK 2048
#define NTHR 256

// physics constants (f32, folded at compile time)
__device__ constexpr float K_P_ATM    = 0.101325f;
__device__ constexpr float K_BETA     = 1466.96f;
__device__ constexpr float K_INV_BETA = 1.0f / 1466.96f;
__device__ constexpr float K_R        = 0.005f / 0.995f;           // AIR_FRACTION/(1-AIR_FRACTION)
__device__ constexpr float K_RP       = (0.005f / 0.995f) * 0.101325f;
__device__ constexpr float K_RHO_MIX  = 851.6f + 1.225f * (0.005f / 0.995f);
__device__ constexpr float K_PT_EPS   = 1e-12f;

__device__ __forceinline__ float wave_reduce_add(float v) {
  #pragma unroll
  for (int off = 16; off >= 1; off >>= 1)
    v += __shfl_xor(v, off, 32);
  return v;
}

__device__ __forceinline__ void accum_elem(float pp, float pt, float md, float tg,
                                           float Vv, float& accP, float& accM) {
  // drho_mix_dp(pp * 0.1), POLYTROPIC_INDEX == 1 so pow collapses to a ratio
  float p      = pp * 0.1f;
  float pu     = fmaxf(p, 1.0f);                   // P_MIN = 1.0
  float inv_pu = 1.0f / pu;
  float pden   = K_RP * inv_pu;                    // r * P_ATM / p_used
  float prat   = pden * inv_pu;                    // pden / (p_used * n)
  float e      = __expf(-(pu - K_P_ATM) * K_INV_BETA) * K_INV_BETA;
  float den    = K_BETA * e + pden;
  float drho   = K_RHO_MIX * (e + prat) / (den * den);
  float res    = Vv * drho * pt - md;
  bool  m      = fabsf(pt) >= K_PT_EPS;
  accP += m ? fabsf(res) : 0.0f;
  accM += m ? fabsf(tg - pp) : 0.0f;
}

// Pass 1: memory-bound stream over B*S elements, b128 loads, per-block partials.
__global__ __launch_bounds__(NTHR) void loss_partials_kernel(
    const float* __restrict__ p_pred,
    const float* __restrict__ p_t_pred,
    const float* __restrict__ mdot_A,
    const float* __restrict__ tgt,
    const float* __restrict__ V,
    float* __restrict__ ws,
    long long n4, long long n)
{
  const float Vv = V[0];   // uniform -> scalar load
  float accP = 0.0f, accM = 0.0f;

  const long long stride = (long long)gridDim.x * blockDim.x;
  for (long long i = (long long)blockIdx.x * blockDim.x + threadIdx.x; i < n4; i += stride) {
    float4 pp = ((const float4*)p_pred)[i];
    float4 pt = ((const float4*)p_t_pred)[i];
    float4 md = ((const float4*)mdot_A)[i];
    float4 tg = ((const float4*)tgt)[i];
    accum_elem(pp.x, pt.x, md.x, tg.x, Vv, accP, accM);
    accum_elem(pp.y, pt.y, md.y, tg.y, Vv, accP, accM);
    accum_elem(pp.z, pt.z, md.z, tg.z, Vv, accP, accM);
    accum_elem(pp.w, pt.w, md.w, tg.w, Vv, accP, accM);
  }
  // scalar tail (n not multiple of 4) — handled by first few threads of block 0
  long long tail0 = n4 * 4;
  if (blockIdx.x == 0) {
    for (long long i = tail0 + threadIdx.x; i < n; i += blockDim.x)
      accum_elem(p_pred[i], p_t_pred[i], mdot_A[i], tgt[i], Vv, accP, accM);
  }

  // block reduction: wave shfl then LDS across 8 waves
  accP = wave_reduce_add(accP);
  accM = wave_reduce_add(accM);
  __shared__ float sP[NTHR / 32], sM[NTHR / 32];
  int lane = threadIdx.x & 31, wid = threadIdx.x >> 5;
  if (lane == 0) { sP[wid] = accP; sM[wid] = accM; }
  __syncthreads();
  if (threadIdx.x == 0) {
    float tP = 0.0f, tM = 0.0f;
    #pragma unroll
    for (int w = 0; w < NTHR / 32; ++w) { tP += sP[w]; tM += sM[w]; }
    ws[blockIdx.x]        = tP;   // physics partials [0, NBLK)
    ws[NBLK + blockIdx.x] = tM;   // M-loss  partials [NBLK, 2*NBLK)
  }
}

// Pass 2: single wave32. WMMA f32 16x16x4 with A=ones reduces 64 partials per
// issue; accumulator C carries the loop (D = ones x B + C). Also computes BCE.
__global__ __launch_bounds__(32) void loss_finalize_kernel(
    const float* __restrict__ ws,
    const float* __restrict__ fault_logit,
    const float* __restrict__ tgt_fault,
    float* __restrict__ out,
    int nB, float inv_bs)
{
  const int lane = threadIdx.x;
  v2f a; a.x = 1.0f; a.y = 1.0f;       // ones A-matrix (16x4 striped across wave)
  v8f cP = {};                          // physics accumulator (C/D)
  v8f cM = {};                          // M-loss accumulator

  #pragma unroll 4
  for (int it = 0; it < NBLK / 64; ++it) {
    int base = it * 64 + lane;
    v2f bP, bM;
    bP.x = ws[base];            bP.y = ws[base + 32];
    bM.x = ws[NBLK + base];     bM.y = ws[NBLK + base + 32];
    // 8 args: (neg_a, A, neg_b, B, c_mod, C, reuse_a, reuse_b)
    cP = __builtin_amdgcn_wmma_f32_16x16x4_f32(false, a, false, bP, (short)0, cP, false, false);
    cM = __builtin_amdgcn_wmma_f32_16x16x4_f32(false, a, false, bM, (short)0, cM, false, false);
  }
  // All D rows equal (A = ones): VGPR0 holds row M=0 (lanes 0-15) and the
  // identical row M=8 (lanes 16-31) -> full-wave sum is 2x the grand total.
  float sumP = 0.5f * wave_reduce_add(cP[0]);
  float sumM = 0.5f * wave_reduce_add(cM[0]);

  // BCE over nB logits
  float accC = 0.0f;
  for (int i = lane; i < nB; i += 32) {
    float x = fault_logit[i];
    float y = tgt_fault[i];
    accC += fmaxf(x, 0.0f) - x * y + log1pf(__expf(-fabsf(x)));
  }
  float sumC = wave_reduce_add(accC);

  if (lane == 0) {
    float lp = sumP * inv_bs;
    float lm = sumM * inv_bs;
    float lc = sumC / (float)nB;
    out[0] = lm + 0.5f * lp + 0.5f * lc;  // total (PHYSICS_WEIGHT = CLS_WEIGHT = 0.5)
    out[1] = lm;
    out[2] = lp;
    out[3] = lc;
  }
}

extern "C" void kernel_launch(void* const* d_in, const int* in_sizes, int n_in,
                              void* d_out, int out_size, void* d_ws, size_t ws_size,
                              hipStream_t stream) {
  const float* p_pred        = (const float*)d_in[0];   // (B,S,1) f32
  const float* p_t_pred      = (const float*)d_in[1];   // (B,S,1) f32
  const float* fault_logit   = (const float*)d_in[2];   // (B,1)   f32
  const float* targets_fault = (const float*)d_in[3];   // (B,)    f32
  const float* mdot_A        = (const float*)d_in[4];   // (B,S)   f32
  const float* targets_p     = (const float*)d_in[5];   // (B,S,1) f32
  const float* V             = (const float*)d_in[6];   // (1,)    f32
  float*       out           = (float*)d_out;
  float*       ws            = (float*)d_ws;            // 2*NBLK floats used

  long long n  = (long long)in_sizes[0];                // B*S
  long long n4 = n / 4;
  int nB       = in_sizes[3];                           // B

  loss_partials_kernel<<<NBLK, NTHR, 0, stream>>>(
      p_pred, p_t_pred, mdot_A, targets_p, V, ws, n4, n);
  loss_finalize_kernel<<<1, 32, 0, stream>>>(
      ws, fault_logit, targets_fault, out, nB, 1.0f / (float)n);
}